// SpectralLossGPU_64020782515056
// MI455X (gfx1250) — compile-verified
//
#include <hip/hip_runtime.h>
#include <hip/hip_bf16.h>

#define G 128
#define GG (G * G)
#define BATCH 8
#define NPTS 524288

typedef __attribute__((ext_vector_type(2))) float v2f;
typedef __attribute__((ext_vector_type(8))) float v8f;

// ---------------------------------------------------------------------------
// 0) zero scratch region (scatter accumulators) — must run every launch
// ---------------------------------------------------------------------------
__global__ void zero_kernel(float* p, long long n) {
    long long i = (long long)blockIdx.x * blockDim.x + threadIdx.x;
    if (i < n) p[i] = 0.0f;
}

// ---------------------------------------------------------------------------
// 1) DFT matrix tables: F[k][m] = exp(-2*pi*i*k*m/G)  (km reduced mod G)
// ---------------------------------------------------------------------------
__global__ void dft_table_kernel(float* Fr, float* Fi) {
    int idx = blockIdx.x * blockDim.x + threadIdx.x;
    if (idx >= GG) return;
    int k = idx >> 7, m = idx & (G - 1);
    int t = (k * m) & (G - 1);
    float ang = -6.2831853071795864769f * (float)t / (float)G;
    Fr[idx] = __cosf(ang);
    Fi[idx] = __sinf(ang);
}

// ---------------------------------------------------------------------------
// 2) scatter points to grid (atomic accumulate), B*N threads
//    field layout: [b][c][GG], c = {pred_r, pred_i, true_r, true_i}
// ---------------------------------------------------------------------------
__global__ void scatter_kernel(const float* __restrict__ y_pred,
                               const float* __restrict__ y_true,
                               const float* __restrict__ coords,
                               float* __restrict__ field,
                               float* __restrict__ count) {
    long long gid = (long long)blockIdx.x * blockDim.x + threadIdx.x;
    if (gid >= (long long)BATCH * NPTS) return;
    int b = (int)(gid / NPTS);
    const float x_min = 0.0f, x_max = 4.5f, y_min = 0.5f, y_max = 5.5f;
    float cx = coords[gid * 3 + 0];
    float cy = coords[gid * 3 + 1];
    float xn = (cx - x_min) / (x_max - x_min) * (float)(G - 1);
    float yn = (cy - y_min) / (y_max - y_min) * (float)(G - 1);
    int xi = (int)rintf(xn);  // round-half-even matches jnp.round
    int yi = (int)rintf(yn);
    xi = min(max(xi, 0), G - 1);
    yi = min(max(yi, 0), G - 1);
    int flat = yi * G + xi;
    float v0 = y_pred[gid * 2 + 0];
    float v1 = y_pred[gid * 2 + 1];
    float v2 = y_true[gid * 2 + 0];
    float v3 = y_true[gid * 2 + 1];
    long long base = (long long)b * 4 * GG + flat;
    atomicAdd(&field[base + 0 * GG], v0);
    atomicAdd(&field[base + 1 * GG], v1);
    atomicAdd(&field[base + 2 * GG], v2);
    atomicAdd(&field[base + 3 * GG], v3);
    atomicAdd(&count[(long long)b * GG + flat], 1.0f);
}

// ---------------------------------------------------------------------------
// 3) field /= count where count > 0
// ---------------------------------------------------------------------------
__global__ void normalize_kernel(float* __restrict__ field,
                                 const float* __restrict__ count) {
    long long idx = (long long)blockIdx.x * blockDim.x + threadIdx.x;
    if (idx >= (long long)BATCH * GG) return;
    float c = count[idx];
    if (c > 0.0f) {
        int b = (int)(idx / GG), cell = (int)(idx % GG);
        float inv = 1.0f / c;
        long long base = (long long)b * 4 * GG + cell;
        field[base + 0 * GG] *= inv;
        field[base + 1 * GG] *= inv;
        field[base + 2 * GG] *= inv;
        field[base + 3 * GG] *= inv;
    }
}

// ---------------------------------------------------------------------------
// 4) one hole-filling iteration (ping-pong):
//    field: 3x3 mean with replicate pad, used where count==0
//    count: 3x3 mean with zero pad; count=1 where (count==0 && csm>0)
// ---------------------------------------------------------------------------
__global__ void holefill_kernel(const float* __restrict__ fin,
                                const float* __restrict__ cin,
                                float* __restrict__ fout,
                                float* __restrict__ cout) {
    long long idx = (long long)blockIdx.x * blockDim.x + threadIdx.x;
    if (idx >= (long long)BATCH * GG) return;
    int b = (int)(idx / GG), cell = (int)(idx % GG);
    int y = cell >> 7, x = cell & (G - 1);
    float c0 = cin[idx];
    float s0 = 0.f, s1 = 0.f, s2 = 0.f, s3 = 0.f, cs = 0.f;
    long long fb = (long long)b * 4 * GG;
    for (int dy = -1; dy <= 1; ++dy) {
        for (int dx = -1; dx <= 1; ++dx) {
            int yy = y + dy, xx = x + dx;
            int yc = min(max(yy, 0), G - 1);
            int xc = min(max(xx, 0), G - 1);
            int ncell = yc * G + xc;
            s0 += fin[fb + 0 * GG + ncell];
            s1 += fin[fb + 1 * GG + ncell];
            s2 += fin[fb + 2 * GG + ncell];
            s3 += fin[fb + 3 * GG + ncell];
            if (yy >= 0 && yy < G && xx >= 0 && xx < G)
                cs += cin[(long long)b * GG + yy * G + xx];
        }
    }
    const float ninth = 1.0f / 9.0f;
    bool hole = (c0 == 0.0f);
    fout[fb + 0 * GG + cell] = hole ? s0 * ninth : fin[fb + 0 * GG + cell];
    fout[fb + 1 * GG + cell] = hole ? s1 * ninth : fin[fb + 1 * GG + cell];
    fout[fb + 2 * GG + cell] = hole ? s2 * ninth : fin[fb + 2 * GG + cell];
    fout[fb + 3 * GG + cell] = hole ? s3 * ninth : fin[fb + 3 * GG + cell];
    cout[idx] = (hole && (cs * ninth > 0.0f)) ? 1.0f : c0;
}

// ---------------------------------------------------------------------------
// 5) complex 128x128 matmul via V_WMMA_F32_16X16X4_F32.
//    O = A * B (complex). One wave (32 lanes) per 16x16 output tile.
//    Per-matrix strides allow A or B to be the shared DFT matrix (stride 0).
//    A layout (16x4 f32): lanes 0-15 hold K=0,1 ; lanes 16-31 hold K=2,3.
//    B layout (4x16 f32): VGPR0 = rows K / K+2 across lane halves, VGPR1 = K+1 / K+3.
//    C/D (16x16 f32): VGPR r -> rows r (lanes 0-15) and r+8 (lanes 16-31).
// ---------------------------------------------------------------------------
__global__ void cmatmul128_wmma_kernel(const float* __restrict__ Ar0,
                                       const float* __restrict__ Ai0,
                                       long long As,
                                       const float* __restrict__ Br0,
                                       const float* __restrict__ Bi0,
                                       long long Bs,
                                       float* __restrict__ Or0,
                                       float* __restrict__ Oi0,
                                       long long Os) {
    int I = blockIdx.x;       // row tile 0..7
    int J = blockIdx.y;       // col tile 0..7
    int m = blockIdx.z;       // matrix index 0..15
    const float* Ar = Ar0 + (long long)m * As;
    const float* Ai = Ai0 + (long long)m * As;
    const float* Br = Br0 + (long long)m * Bs;
    const float* Bi = Bi0 + (long long)m * Bs;
    float* Or = Or0 + (long long)m * Os;
    float* Oi = Oi0 + (long long)m * Os;

    int lane = threadIdx.x & 31;
    int half = lane >> 4;       // 0: K, K+1 ; 1: K+2, K+3
    int l = lane & 15;

    v8f rr = {}, ii = {}, ri = {}, ir = {};
    int arow = I * 16 + l;
    int bcol = J * 16 + l;

    for (int k = 0; k < G; k += 4) {
        int ka = k + half * 2;
        v2f ar = { Ar[arow * G + ka], Ar[arow * G + ka + 1] };
        v2f ai = { Ai[arow * G + ka], Ai[arow * G + ka + 1] };
        v2f br = { Br[ka * G + bcol], Br[(ka + 1) * G + bcol] };
        v2f bi = { Bi[ka * G + bcol], Bi[(ka + 1) * G + bcol] };
        rr = __builtin_amdgcn_wmma_f32_16x16x4_f32(false, ar, false, br,
                                                   (short)0, rr, false, false);
        ii = __builtin_amdgcn_wmma_f32_16x16x4_f32(false, ai, false, bi,
                                                   (short)0, ii, false, false);
        ri = __builtin_amdgcn_wmma_f32_16x16x4_f32(false, ar, false, bi,
                                                   (short)0, ri, false, false);
        ir = __builtin_amdgcn_wmma_f32_16x16x4_f32(false, ai, false, br,
                                                   (short)0, ir, false, false);
    }

    for (int r = 0; r < 8; ++r) {
        int row = I * 16 + r + half * 8;
        int col = J * 16 + l;
        Or[row * G + col] = rr[r] - ii[r];   // real = Ar*Br - Ai*Bi
        Oi[row * G + col] = ri[r] + ir[r];   // imag = Ar*Bi + Ai*Br
    }
}

// ---------------------------------------------------------------------------
// 6) per-batch spectral loss. Y layout: [m][2][GG], m = 2*b + {0:pred,1:true}
//    fftshift skipped: loss is pointwise + permutation-invariant reductions.
// ---------------------------------------------------------------------------
__global__ void loss_kernel(const float* __restrict__ Y, float* __restrict__ ps) {
    int b = blockIdx.x;
    int t = threadIdx.x;  // 256 threads
    const float* Pr = Y + (long long)(2 * b) * 2 * GG;
    const float* Pi = Pr + GG;
    const float* Tr = Y + (long long)(2 * b + 1) * 2 * GG;
    const float* Ti = Tr + GG;
    const float scale = (4.5f / 127.0f) * (5.0f / 127.0f);  // dx*dy

    __shared__ float red[256];

    float fmaxv = 0.0f;
    for (int i = t; i < GG; i += 256) {
        float fr = Tr[i] * scale, fi = Ti[i] * scale;
        fmaxv = fmaxf(fmaxv, sqrtf(fr * fr + fi * fi));
    }
    red[t] = fmaxv;
    __syncthreads();
    for (int s = 128; s > 0; s >>= 1) {
        if (t < s) red[t] = fmaxf(red[t], red[t + s]);
        __syncthreads();
    }
    fmaxv = red[0];
    __syncthreads();

    float thresh = fmaxv / 1000.0f;
    float denomw = fmaxv * fmaxv + 1e-10f;
    float wsum = 0.0f, msum = 0.0f;
    for (int i = t; i < GG; i += 256) {
        float tr = Tr[i] * scale, ti = Ti[i] * scale;
        float fa = sqrtf(tr * tr + ti * ti);
        float mask = (fa >= thresh) ? 1.0f : 0.0f;
        float w = fa * fa / denomw;
        float dr = (Pr[i] * scale - tr) * mask;
        float di = (Pi[i] * scale - ti) * mask;
        wsum += (dr * dr + di * di) * w;
        msum += mask;
    }
    red[t] = wsum;
    __syncthreads();
    for (int s = 128; s > 0; s >>= 1) {
        if (t < s) red[t] += red[t + s];
        __syncthreads();
    }
    wsum = red[0];
    __syncthreads();
    red[t] = msum;
    __syncthreads();
    for (int s = 128; s > 0; s >>= 1) {
        if (t < s) red[t] += red[t + s];
        __syncthreads();
    }
    if (t == 0) ps[b] = wsum / (red[0] + 1e-10f);
}

__global__ void mean_kernel(const float* __restrict__ ps, float* __restrict__ out) {
    if (threadIdx.x == 0) {
        float s = 0.0f;
        for (int b = 0; b < BATCH; ++b) s += ps[b];
        out[0] = s / (float)BATCH;
    }
}

// ---------------------------------------------------------------------------
// launch
// ---------------------------------------------------------------------------
extern "C" void kernel_launch(void* const* d_in, const int* in_sizes, int n_in,
                              void* d_out, int out_size, void* d_ws, size_t ws_size,
                              hipStream_t stream) {
    const float* y_pred = (const float*)d_in[0];
    const float* y_true = (const float*)d_in[1];
    const float* coords = (const float*)d_in[2];
    float* out = (float*)d_out;

    // workspace layout (floats)
    float* ws = (float*)d_ws;
    float* fieldA = ws;                       // B*4*GG = 524288
    float* countA = fieldA + (long long)BATCH * 4 * GG;  // B*GG = 131072
    float* fieldB = countA + (long long)BATCH * GG;      // 524288
    float* countB = fieldB + (long long)BATCH * 4 * GG;  // 131072
    float* Fr     = countB + (long long)BATCH * GG;      // GG
    float* Fi     = Fr + GG;                             // GG
    float* T      = Fi + GG;                             // 16*2*GG = 524288
    float* Y      = T + (long long)16 * 2 * GG;          // 524288
    float* ps     = Y + (long long)16 * 2 * GG;          // 8

    // 0) zero scatter accumulators (fieldA + countA are contiguous)
    {
        long long n = (long long)BATCH * 4 * GG + (long long)BATCH * GG;
        zero_kernel<<<(int)((n + 255) / 256), 256, 0, stream>>>(fieldA, n);
    }
    // 1) DFT tables
    dft_table_kernel<<<GG / 256, 256, 0, stream>>>(Fr, Fi);
    // 2) scatter
    {
        long long n = (long long)BATCH * NPTS;
        scatter_kernel<<<(int)((n + 255) / 256), 256, 0, stream>>>(
            y_pred, y_true, coords, fieldA, countA);
    }
    // 3) normalize by count
    {
        int n = BATCH * GG;
        normalize_kernel<<<(n + 255) / 256, 256, 0, stream>>>(fieldA, countA);
    }
    // 4) 3 hole-fill iterations: A->B->A->B (final in B)
    {
        int n = BATCH * GG;
        int g = (n + 255) / 256;
        holefill_kernel<<<g, 256, 0, stream>>>(fieldA, countA, fieldB, countB);
        holefill_kernel<<<g, 256, 0, stream>>>(fieldB, countB, fieldA, countA);
        holefill_kernel<<<g, 256, 0, stream>>>(fieldA, countA, fieldB, countB);
    }
    // 5) FFT2 = F * X * F per complex matrix (16 matrices: [b][pred/true])
    //    fieldB layout [b][4][GG] => complex matrix m=2b+f at offset 2m*GG (re), +GG (im)
    {
        dim3 grid(8, 8, 16), block(32);
        // stage 1: T = F * X   (A = shared DFT, stride 0; B = fields, stride 2*GG)
        cmatmul128_wmma_kernel<<<grid, block, 0, stream>>>(
            Fr, Fi, 0LL, fieldB, fieldB + GG, (long long)2 * GG,
            T, T + GG, (long long)2 * GG);
        // stage 2: Y = T * F
        cmatmul128_wmma_kernel<<<grid, block, 0, stream>>>(
            T, T + GG, (long long)2 * GG, Fr, Fi, 0LL,
            Y, Y + GG, (long long)2 * GG);
    }
    // 6) per-batch loss + mean
    loss_kernel<<<BATCH, 256, 0, stream>>>(Y, ps);
    mean_kernel<<<1, 32, 0, stream>>>(ps, out);
}